// SpatialRelationHead_63642825392171
// MI455X (gfx1250) — compile-verified
//
#include <hip/hip_runtime.h>

typedef __attribute__((ext_vector_type(2))) float v2f;
typedef __attribute__((ext_vector_type(4))) float v4f;
typedef __attribute__((ext_vector_type(8))) float v8f;

#define CCH   64            // channels
#define NPIX  6400          // 80*80
#define NB    2             // batch
#define MS    4             // m-range split (parallelism over the reduction dim)
#define MTPS  100           // m-tiles per split: (NPIX/16)/MS
#define LDSW  20            // padded LDS row stride (dwords): 16B-aligned, conflict-free

// ---------------------------------------------------------------------------
// Kernel 1: g_x[b][n][i] = sum_c g_w[i][c] * x[b][c][n] + g_b[i]   (layout b,n,i)
// ---------------------------------------------------------------------------
__global__ void __launch_bounds__(256) gx_kernel(const float* __restrict__ x,
                                                 const float* __restrict__ g_w,
                                                 const float* __restrict__ g_b,
                                                 float* __restrict__ g_x) {
    __shared__ float w_lds[64 * 65];
    int bz    = blockIdx.x / 50;
    int nbase = (blockIdx.x % 50) * 128;
    for (int idx = threadIdx.x; idx < 64 * 64; idx += 256)
        w_lds[(idx >> 6) * 65 + (idx & 63)] = g_w[idx];
    __syncthreads();
    const float* xb = x + (size_t)bz * CCH * NPIX;
    int i = threadIdx.x & 63;
    float bias = g_b[i];
    for (int it = 0; it < 32; ++it) {
        int idx = threadIdx.x + 256 * it;
        int n   = nbase + (idx >> 6);
        float acc = bias;
        #pragma unroll 8
        for (int c = 0; c < 64; ++c)
            acc += w_lds[i * 65 + c] * xb[(size_t)c * NPIX + n];
        g_x[((size_t)bz * NPIX + n) * CCH + i] = acc;
    }
}

// ---------------------------------------------------------------------------
// Kernel 2 (main, fused): wave = (batch, 16-row n-tile, m-split s).
//   for mt in s-th quarter of 400 m-tiles:
//     aff(16x16)  = Xt[n-tile] * X[m-tile]   -> 16 WMMA, two independent chains
//     score       = sigmoid(aff)             -> staged in LDS, b128 streamed out
//     emb(16x64) += score * g_x[m-tile]      -> 16 WMMA into 4 resident chains
//   emb partial written to emb_part[s] (summed deterministically in kernel 3).
// ---------------------------------------------------------------------------
__global__ void __launch_bounds__(64) fused_attn_kernel(const float* __restrict__ x,
                                                        const float* __restrict__ g_x,
                                                        float* __restrict__ score,
                                                        float* __restrict__ emb_part) {
    __shared__ float tile[2][16 * LDSW];
    const int wave = threadIdx.x >> 5;
    const int lane = threadIdx.x & 31;
    const int hi   = lane >> 4;
    const int lc   = lane & 15;

    const int bz   = blockIdx.x / (MS * 200);
    const int rem  = blockIdx.x % (MS * 200);
    const int s    = rem / 200;
    const int pair = rem % 200;
    const int n0   = (pair * 2 + wave) * 16;

    const float* xb  = x   + (size_t)bz * CCH * NPIX;
    const float* gxb = g_x + (size_t)bz * NPIX * CCH;
    float* my = &tile[wave][0];

    // Loop-invariant A operand: A[r][k] = x[k][n0+r]
    v2f A[16];
    #pragma unroll
    for (int kk = 0; kk < 16; ++kk) {
        int k = 4 * kk + 2 * hi;
        A[kk].x = xb[(size_t)k * NPIX + n0 + lc];
        A[kk].y = xb[(size_t)(k + 1) * NPIX + n0 + lc];
    }

    v8f acc0 = {}, acc1 = {}, acc2 = {}, acc3 = {};   // resident emb tile 16x64

    const int r   = lane >> 1;          // row for wide score store
    const int cb8 = (lane & 1) * 8;     // col base (8 floats per lane)

    for (int mt = s * MTPS; mt < (s + 1) * MTPS; ++mt) {
        const int m0 = mt * 16;
        // ---- aff tile: K=64, two independent WMMA accumulator chains ----
        v8f ca = {}, cb = {};
        #pragma unroll
        for (int kk = 0; kk < 16; kk += 2) {
            int k0 = 4 * kk + 2 * hi;
            int k1 = k0 + 4;
            v2f B0, B1;
            B0.x = xb[(size_t)k0 * NPIX + m0 + lc];
            B0.y = xb[(size_t)(k0 + 1) * NPIX + m0 + lc];
            B1.x = xb[(size_t)k1 * NPIX + m0 + lc];
            B1.y = xb[(size_t)(k1 + 1) * NPIX + m0 + lc];
            ca = __builtin_amdgcn_wmma_f32_16x16x4_f32(false, A[kk],     false, B0, (short)0, ca, false, false);
            cb = __builtin_amdgcn_wmma_f32_16x16x4_f32(false, A[kk + 1], false, B1, (short)0, cb, false, false);
        }
        v8f cacc = ca + cb;
        // ---- sigmoid, stage tile in LDS (DS in-order per wave) ----
        #pragma unroll
        for (int j = 0; j < 8; ++j) {
            float v = 1.0f / (1.0f + __expf(-cacc[j]));
            my[(j + 8 * hi) * LDSW + lc] = v;        // C/D layout: vgpr j -> row j+8*hi
        }
        // ---- wide score store: 2 x b128 per lane, rows contiguous ----
        {
            v4f q0 = *(const v4f*)(my + r * LDSW + cb8);
            v4f q1 = *(const v4f*)(my + r * LDSW + cb8 + 4);
            float* sp = score + ((size_t)(bz * NPIX) + n0 + r) * NPIX + m0 + cb8;
            *(v4f*)sp       = q0;
            *(v4f*)(sp + 4) = q1;
        }
        // ---- emb tile += score(16x16) * g_x[m-tile](16x64), K=16 ----
        #pragma unroll
        for (int kc = 0; kc < 4; ++kc) {
            int kb = 4 * kc + 2 * hi;
            v2f SA = *(const v2f*)(my + lc * LDSW + kb);   // A-layout read of score tile
            const float* g0 = gxb + (size_t)(m0 + kb) * CCH;
            const float* g1 = g0 + CCH;
            v2f B0, B1, B2, B3;
            B0.x = g0[lc];      B0.y = g1[lc];
            B1.x = g0[16 + lc]; B1.y = g1[16 + lc];
            B2.x = g0[32 + lc]; B2.y = g1[32 + lc];
            B3.x = g0[48 + lc]; B3.y = g1[48 + lc];
            acc0 = __builtin_amdgcn_wmma_f32_16x16x4_f32(false, SA, false, B0, (short)0, acc0, false, false);
            acc1 = __builtin_amdgcn_wmma_f32_16x16x4_f32(false, SA, false, B1, (short)0, acc1, false, false);
            acc2 = __builtin_amdgcn_wmma_f32_16x16x4_f32(false, SA, false, B2, (short)0, acc2, false, false);
            acc3 = __builtin_amdgcn_wmma_f32_16x16x4_f32(false, SA, false, B3, (short)0, acc3, false, false);
        }
    }

    // ---- write emb partial, layout [s][b][n][i] ----
    #pragma unroll
    for (int j = 0; j < 8; ++j) {
        int n = n0 + j + 8 * hi;
        size_t base = (((size_t)s * NB + bz) * NPIX + n) * CCH;
        emb_part[base + lc]      = acc0[j];
        emb_part[base + 16 + lc] = acc1[j];
        emb_part[base + 32 + lc] = acc2[j];
        emb_part[base + 48 + lc] = acc3[j];
    }
}

// ---------------------------------------------------------------------------
// Kernel 3: emb = sum_s emb_part[s] (fixed order); y = w_w*emb + w_b (b,o,n);
// per-(block,group) sum/sumsq via deterministic tree (no atomics).
// ---------------------------------------------------------------------------
__global__ void __launch_bounds__(256) y_stats_kernel(const float* __restrict__ emb_part,
                                                      const float* __restrict__ w_w,
                                                      const float* __restrict__ w_b,
                                                      float* __restrict__ y_ws,
                                                      float* __restrict__ part_stats) {
    __shared__ float w_lds[64 * 65];
    __shared__ float rsum[256], rsq[256];
    const size_t PART = (size_t)NB * NPIX * CCH;
    int bz    = blockIdx.x / 50;
    int nbase = (blockIdx.x % 50) * 128;
    for (int idx = threadIdx.x; idx < 64 * 64; idx += 256)
        w_lds[(idx >> 6) * 65 + (idx & 63)] = w_w[idx];
    __syncthreads();
    int o = threadIdx.x & 63;
    float bias = w_b[o], lsum = 0.f, lsq = 0.f;
    for (int it = 0; it < 32; ++it) {
        int idx = threadIdx.x + 256 * it;
        int n   = nbase + (idx >> 6);
        const float* e = emb_part + ((size_t)bz * NPIX + n) * CCH;
        float a = bias;
        #pragma unroll 8
        for (int c = 0; c < 64; ++c) {
            float ev = e[c] + e[PART + c] + e[2 * PART + c] + e[3 * PART + c];
            a += w_lds[o * 65 + c] * ev;
        }
        y_ws[((size_t)bz * 64 + o) * NPIX + n] = a;
        lsum += a; lsq += a * a;
    }
    rsum[threadIdx.x] = lsum;
    rsq[threadIdx.x]  = lsq;
    __syncthreads();
    if (threadIdx.x < 8) {                      // fixed-order per-group reduction
        int g = threadIdx.x;
        float ssum = 0.f, ssq = 0.f;
        for (int k = 0; k < 4; ++k)
            for (int oo = 0; oo < 8; ++oo) {
                int t = (8 * g + oo) + 64 * k;
                ssum += rsum[t]; ssq += rsq[t];
            }
        part_stats[(blockIdx.x * 8 + g) * 2 + 0] = ssum;
        part_stats[(blockIdx.x * 8 + g) * 2 + 1] = ssq;
    }
}

// Fixed-order reduction of 50 per-block partials -> stats[2*(b*8+g)+{0,1}]
__global__ void stats_reduce_kernel(const float* __restrict__ part_stats,
                                    float* __restrict__ stats) {
    int t = threadIdx.x;
    if (t >= 32) return;
    int b = t >> 4, g = (t >> 1) & 7, w = t & 1;
    float s = 0.f;
    for (int nb = 0; nb < 50; ++nb)
        s += part_stats[(((b * 50) + nb) * 8 + g) * 2 + w];
    stats[2 * (b * 8 + g) + w] = s;
}

// ---------------------------------------------------------------------------
// Kernel 4: GroupNorm normalize + affine -> out
// ---------------------------------------------------------------------------
__global__ void __launch_bounds__(256) norm_kernel(const float* __restrict__ y_ws,
                                                   const float* __restrict__ stats,
                                                   const float* __restrict__ gn_w,
                                                   const float* __restrict__ gn_b,
                                                   float* __restrict__ out) {
    int idx = blockIdx.x * 256 + threadIdx.x;
    if (idx >= NB * 64 * NPIX) return;
    int bz  = idx / (64 * NPIX);
    int o   = (idx % (64 * NPIX)) / NPIX;
    int g   = o >> 3;
    const float cnt = 8.0f * (float)NPIX;
    float mean = stats[2 * (bz * 8 + g)] / cnt;
    float var  = stats[2 * (bz * 8 + g) + 1] / cnt - mean * mean;
    float inv  = rsqrtf(var + 1e-5f);
    out[idx] = (y_ws[idx] - mean) * inv * gn_w[o] + gn_b[o];
}

// ---------------------------------------------------------------------------
extern "C" void kernel_launch(void* const* d_in, const int* in_sizes, int n_in,
                              void* d_out, int out_size, void* d_ws, size_t ws_size,
                              hipStream_t stream) {
    const float* x    = (const float*)d_in[0];
    const float* g_w  = (const float*)d_in[1];
    const float* g_b  = (const float*)d_in[2];
    const float* w_w  = (const float*)d_in[3];
    const float* w_b  = (const float*)d_in[4];
    const float* gn_w = (const float*)d_in[5];
    const float* gn_b = (const float*)d_in[6];

    float* out   = (float*)d_out;                              // 819200 floats
    float* score = (float*)d_out + (size_t)NB * CCH * NPIX;    // 81.92M floats

    const size_t SEG = (size_t)NB * NPIX * CCH;                // 819200
    float* ws         = (float*)d_ws;
    float* g_x        = ws;                                    // SEG
    float* emb_part   = ws + SEG;                              // MS*SEG
    float* y_ws       = ws + (1 + MS) * SEG;                   // SEG
    float* part_stats = ws + (2 + MS) * SEG;                   // 1600
    float* stats      = part_stats + 1600;                     // 32

    gx_kernel<<<100, 256, 0, stream>>>(x, g_w, g_b, g_x);
    fused_attn_kernel<<<NB * MS * 200, 64, 0, stream>>>(x, g_x, score, emb_part);
    y_stats_kernel<<<100, 256, 0, stream>>>(emb_part, w_w, w_b, y_ws, part_stats);
    stats_reduce_kernel<<<1, 32, 0, stream>>>(part_stats, stats);
    norm_kernel<<<(NB * 64 * NPIX + 255) / 256, 256, 0, stream>>>(y_ws, stats, gn_w, gn_b, out);
}